// BKTNN_76596446757102
// MI455X (gfx1250) — compile-verified
//
#include <hip/hip_runtime.h>

// Problem constants (from the reference).
#define B_N 65536
#define T_N 200
#define H_N 512
#define S_N 1000
#define P_N 5
#define EPSF 1e-6f

typedef __attribute__((ext_vector_type(2))) float v2f;
typedef __attribute__((ext_vector_type(8))) float v8f;

// ---------------------------------------------------------------------------
// Async global->LDS copies (CDNA5 ASYNCcnt path, CDNA5 ISA).
// GVS addressing: mem = SGPR_base + VGPR_off32 + inst_offset.
// VDST operand is the per-lane LDS byte address; low 32 bits of a generic
// pointer to __shared__ are the wave-relative LDS address (ISA §10.2).
// ---------------------------------------------------------------------------
__device__ __forceinline__ void async_g2lds_b32(unsigned lds_addr, unsigned goff,
                                                unsigned long long gbase) {
    asm volatile("global_load_async_to_lds_b32 %0, %1, %2"
                 :: "v"(lds_addr), "v"(goff), "s"(gbase) : "memory");
}
__device__ __forceinline__ void async_g2lds_b128(unsigned lds_addr, unsigned goff,
                                                 unsigned long long gbase) {
    asm volatile("global_load_async_to_lds_b128 %0, %1, %2"
                 :: "v"(lds_addr), "v"(goff), "s"(gbase) : "memory");
}
__device__ __forceinline__ void wait_async0() {
    asm volatile("s_wait_asynccnt 0x0" ::: "memory");
}

// ---------------------------------------------------------------------------
// Kernel 1: params_all[s][p] for ALL S skills (embed[X] has only S distinct
// rows, so the 2x(HxH) MLP is hoisted from B=65536 to S=1000 rows: 68.7 GFLOP
// -> 1.05 GFLOP, bit-identical per-row math).
// fp32 WMMA (V_WMMA_F32_16X16X4_F32): M=16 rows/workgroup, 8 waves x 4
// N-tiles, layers computed in-place in one padded LDS tile.
// ---------------------------------------------------------------------------
__global__ __launch_bounds__(256) void bkt_mlp_params_kernel(
    const float* __restrict__ embed,
    const float* __restrict__ W1, const float* __restrict__ b1,
    const float* __restrict__ W2, const float* __restrict__ b2,
    const float* __restrict__ Wout, const float* __restrict__ bout,
    float* __restrict__ params_all)
{
    constexpr int LDSTR = H_N + 4;           // 516 floats: bank padding; 2064 B row (16B-aligned)
    __shared__ float buf[16 * LDSTR];        // 33 KB of the 320 KB WGP LDS

    const int tid  = threadIdx.x;
    const int lane = tid & 31;               // wave32
    const int wave = tid >> 5;               // 8 waves
    const int half = lane >> 4;              // A/B K-half select (ISA 7.12.2)
    const int lrow = lane & 15;              // A row / B,C,D column
    const int s0   = blockIdx.x * 16;

    // Stage 16 embedding rows via async global->LDS, 16 B per lane.
    for (int idx = tid; idx < 16 * (H_N / 4); idx += 256) {
        const int r  = idx >> 7;             // / (H_N/4)
        const int c4 = idx & (H_N / 4 - 1);
        int s = s0 + r; if (s >= S_N) s = S_N - 1;
        const unsigned lds_addr = (unsigned)(size_t)&buf[r * LDSTR + 4 * c4];
        const unsigned goff     = (unsigned)(((size_t)s * H_N + 4 * c4) * sizeof(float));
        async_g2lds_b128(lds_addr, goff, (unsigned long long)embed);
    }
    wait_async0();
    __syncthreads();

    // Two hidden layers, in-place.
    const float* Ws[2] = { W1, W2 };
    const float* bs[2] = { b1, b2 };
    for (int layer = 0; layer < 2; ++layer) {
        const float* __restrict__ W    = Ws[layer];
        const float* __restrict__ bias = bs[layer];
        v8f acc[4];
        for (int i = 0; i < 4; ++i) {
            const int n0 = (wave + 8 * i) * 16;
            v8f c = {};
            for (int k0 = 0; k0 < H_N; k0 += 4) {
                const int ka = k0 + 2 * half;
                v2f av, bv;
                // A 16x4 f32: lanes 0-15 hold K=0,1; lanes 16-31 hold K=2,3
                av.x = buf[lrow * LDSTR + ka];
                av.y = buf[lrow * LDSTR + ka + 1];
                // B 4x16 f32: lane -> N, VGPR/half -> K (mirror of A layout)
                bv.x = W[(size_t)ka * H_N + n0 + lrow];
                bv.y = W[(size_t)(ka + 1) * H_N + n0 + lrow];
                c = __builtin_amdgcn_wmma_f32_16x16x4_f32(
                        false, av, false, bv, (short)0, c, false, false);
            }
            acc[i] = c;
        }
        __syncthreads();                     // all waves done READING buf
        for (int i = 0; i < 4; ++i) {
            const int n0 = (wave + 8 * i) * 16;
            const float bb = bias[n0 + lrow];
#pragma unroll
            for (int r = 0; r < 8; ++r) {    // C/D: VGPR r -> row r + 8*half
                float v = acc[i][r] + bb;
                buf[(r + 8 * half) * LDSTR + n0 + lrow] = fmaxf(v, 0.0f);
            }
        }
        __syncthreads();
    }

    // Output head: single 16-wide N tile (P=5 valid columns), wave 0 only.
    if (wave == 0) {
        v8f c = {};
        for (int k0 = 0; k0 < H_N; k0 += 4) {
            const int ka = k0 + 2 * half;
            v2f av, bv;
            av.x = buf[lrow * LDSTR + ka];
            av.y = buf[lrow * LDSTR + ka + 1];
            bv.x = (lrow < P_N) ? Wout[(size_t)ka * P_N + lrow] : 0.0f;
            bv.y = (lrow < P_N) ? Wout[(size_t)(ka + 1) * P_N + lrow] : 0.0f;
            c = __builtin_amdgcn_wmma_f32_16x16x4_f32(
                    false, av, false, bv, (short)0, c, false, false);
        }
        const float bb = (lrow < P_N) ? bout[lrow] : 0.0f;
#pragma unroll
        for (int r = 0; r < 8; ++r) {
            const int row = r + 8 * half;
            const int s   = s0 + row;
            if (s < S_N && lrow < P_N) {
                float v = c[r] + bb;
                v = 1.0f / (1.0f + __expf(-v));              // sigmoid
                v = fminf(fmaxf(v, EPSF), 1.0f - EPSF);      // clip
                params_all[s * P_N + lrow] = v;
            }
        }
    }
}

// ---------------------------------------------------------------------------
// Kernel 2: per-student BKT scan over T=200 steps (bandwidth-bound, ~160 MB).
// y/corrects/latents are (B,T) row-major -> stage 8-step chunks through
// padded LDS so all global transactions are contiguous. The y chunk is
// fetched with async global->LDS (no VGPR round-trip); ASYNCcnt drain +
// barrier makes it visible to every wave.
// ---------------------------------------------------------------------------
__global__ __launch_bounds__(256) void bkt_scan_kernel(
    const int* __restrict__ y, const int* __restrict__ X,
    const float* __restrict__ params_all,
    float* __restrict__ corrects, float* __restrict__ latents,
    float* __restrict__ params_out)
{
    constexpr int C  = 8;        // timestep chunk (200 = 25 * 8)
    constexpr int CP = C + 1;    // LDS pad: stride 9 words, 9 invertible mod 64
    __shared__ int   ys[256 * CP];
    __shared__ float cs[256 * CP];
    __shared__ float ls[256 * CP];

    const int tid = threadIdx.x;
    const int b0  = blockIdx.x * 256;
    const int b   = b0 + tid;

    // Gather this student's BKT parameters from the skill table.
    const int sk = X[b];
    float p[P_N];
#pragma unroll
    for (int j = 0; j < P_N; ++j) {
        p[j] = params_all[sk * P_N + j];
        params_out[(size_t)b * P_N + j] = p[j];
    }
    const float l = p[0], g = p[2], s = p[3];
    float latent = p[4];

    for (int t0 = 0; t0 < T_N; t0 += C) {
        // Async global->LDS copy of the y chunk (each 8 lanes cover 32
        // contiguous bytes of one student's row).
        for (int idx = tid; idx < 256 * C; idx += 256) {
            const int r = idx >> 3;
            const int t = idx & (C - 1);
            const unsigned lds_addr = (unsigned)(size_t)&ys[r * CP + t];
            const unsigned goff =
                (unsigned)(((size_t)(b0 + r) * T_N + t0 + t) * sizeof(int));
            async_g2lds_b32(lds_addr, goff, (unsigned long long)y);
        }
        wait_async0();
        __syncthreads();

#pragma unroll
        for (int t = 0; t < C; ++t) {
            // y==-1 maps to 0 in the reference; branch on y > 0.
            const float yt      = (ys[tid * CP + t] > 0) ? 1.0f : 0.0f;
            const float num1    = latent * (1.0f - s);
            const float correct = num1 + (1.0f - latent) * g;
            const float k1      = num1 / correct;
            const float d0      = latent * s + (1.0f - latent) * (1.0f - g);
            const float k0      = latent * s / d0;
            const float k       = (yt > 0.5f) ? k1 : k0;
            cs[tid * CP + t] = correct;
            ls[tid * CP + t] = latent;
            const float nxt = k + (1.0f - k) * l;
            latent = fminf(fmaxf(nxt, EPSF), 1.0f - EPSF);
        }
        __syncthreads();

        // Coalesced store of the chunk's outputs.
        for (int idx = tid; idx < 256 * C; idx += 256) {
            const int r = idx >> 3;
            const int t = idx & (C - 1);
            const size_t o = (size_t)(b0 + r) * T_N + t0 + t;
            corrects[o] = cs[r * CP + t];
            latents[o]  = ls[r * CP + t];
        }
        __syncthreads();
    }
}

// ---------------------------------------------------------------------------
// Launch: inputs in setup_inputs() order:
//   0:X 1:y 2:embed 3:W1 4:b1 5:W2 6:b2 7:Wout 8:bout
// Output: corrects (B*T) | latents (B*T) | params (B*P), all fp32.
// ---------------------------------------------------------------------------
extern "C" void kernel_launch(void* const* d_in, const int* in_sizes, int n_in,
                              void* d_out, int out_size, void* d_ws, size_t ws_size,
                              hipStream_t stream) {
    const int*   X     = (const int*)  d_in[0];
    const int*   y     = (const int*)  d_in[1];
    const float* embed = (const float*)d_in[2];
    const float* W1    = (const float*)d_in[3];
    const float* b1    = (const float*)d_in[4];
    const float* W2    = (const float*)d_in[5];
    const float* b2    = (const float*)d_in[6];
    const float* Wout  = (const float*)d_in[7];
    const float* bout  = (const float*)d_in[8];

    float* out        = (float*)d_out;
    float* corrects   = out;
    float* latents    = out + (size_t)B_N * T_N;
    float* params_out = out + 2 * (size_t)B_N * T_N;

    float* params_all = (float*)d_ws;    // S*P = 20 KB scratch, rewritten every call

    const int mlp_blocks = (S_N + 15) / 16;   // 63
    bkt_mlp_params_kernel<<<mlp_blocks, 256, 0, stream>>>(
        embed, W1, b1, W2, b2, Wout, bout, params_all);

    bkt_scan_kernel<<<B_N / 256, 256, 0, stream>>>(
        y, X, params_all, corrects, latents, params_out);
}